// NeighborEncoderBlock_70557722739516
// MI455X (gfx1250) — compile-verified
//
#include <hip/hip_runtime.h>

typedef unsigned long long u64;

#define Bn 4
#define Cn 5
#define Hn 64
#define Wn 2048
#define Ln (Hn * Wn)
#define SEARCH 5
#define PADR 2
#define KNN 9
#define OUTC 32
#define FEATS 18
#define KPAD 20
#define LEAK 0.01f

typedef __attribute__((ext_vector_type(2))) float v2f;
typedef __attribute__((ext_vector_type(4))) float v4f;
typedef __attribute__((ext_vector_type(8))) float v8f;

// 25 -> 9 smallest-distance selection with a single u64 sort key:
//   key = (f32bits(distance) << 32) | neighbor_index
// distance >= 0 so float-bit ordering is monotonic; the index in the low bits
// reproduces jax.lax.top_k's lower-index-first tie-breaking and makes all keys
// unique. Swap-down insertion keeps keys[] sorted ascending:
// 1 v_cmp_lt_u64 + 4 v_cndmask per slot per candidate.
__device__ __forceinline__ void knn_keys(const float* __restrict__ xb,
                                         const int* __restrict__ mb,
                                         int y, int xx, u64 keys[KNN]) {
  const int l = y * Wn + xx;
  const float c1 = xb[1 * Ln + l];
  const float c2 = xb[2 * Ln + l];
  const float c3 = xb[3 * Ln + l];
#pragma unroll
  for (int k = 0; k < KNN; ++k) keys[k] = ~0ull;
#pragma unroll
  for (int dy = 0; dy < SEARCH; ++dy) {
#pragma unroll
    for (int dx = 0; dx < SEARCH; ++dx) {
      const int i = dy * SEARCH + dx;
      const int ny = y + dy - PADR;
      const int nx = xx + dx - PADR;
      float n1 = 0.f, n2 = 0.f, n3 = 0.f;
      if (ny >= 0 && ny < Hn && nx >= 0 && nx < Wn) {
        const int nl = ny * Wn + nx;
        if (mb[nl] == 0) {  // ~binary_mask
          n1 = xb[1 * Ln + nl];
          n2 = xb[2 * Ln + nl];
          n3 = xb[3 * Ln + nl];
        }
      }
      const float d1 = c1 - n1, d2 = c2 - n2, d3 = c3 - n3;
      const float d = sqrtf(d1 * d1 + d2 * d2 + d3 * d3);
      u64 cur = ((u64)__float_as_uint(d) << 32) | (u64)i;
#pragma unroll
      for (int j = 0; j < KNN; ++j) {
        const u64 old = keys[j];
        const bool c = cur < old;
        keys[j] = c ? cur : old;
        cur = c ? old : cur;
      }
    }
  }
}

// Re-gather the masked channel-0 value for one surviving neighbor index.
__device__ __forceinline__ float gather_g(const float* __restrict__ xb,
                                          const int* __restrict__ mb,
                                          int y, int xx, int idx) {
  const int dy = idx / SEARCH;
  const int dx = idx - dy * SEARCH;
  const int ny = y + dy - PADR;
  const int nx = xx + dx - PADR;
  float g = 0.f;
  if (ny >= 0 && ny < Hn && nx >= 0 && nx < Wn) {
    const int nl = ny * Wn + nx;
    if (mb[nl] == 0) g = xb[nl];
  }
  return g;
}

__global__ void __launch_bounds__(256) init_gmax_kernel(unsigned* g) { *g = 0u; }

// Pass A: write knn_values, reduce global max(g) (float-as-uint, g >= 0),
// optionally persist sorted keys to workspace so pass B can skip the search.
template <bool STORE_KEYS>
__global__ void __launch_bounds__(256) knn_pass_kernel(const float* __restrict__ x,
                                                       const int* __restrict__ mask,
                                                       float* __restrict__ knn_out,
                                                       unsigned* __restrict__ gmax,
                                                       u64* __restrict__ key_ws) {
  const int t = blockIdx.x * blockDim.x + threadIdx.x;
  const int b = t / Ln;
  const int l = t - b * Ln;
  const int y = l / Wn;
  const int xx = l - y * Wn;
  const float* xb = x + (size_t)b * Cn * Ln;
  const int*   mb = mask + (size_t)b * Ln;

  u64 keys[KNN];
  knn_keys(xb, mb, y, xx, keys);

  float* o = knn_out + (size_t)b * KNN * Ln + l;
  float m = 0.f;
#pragma unroll
  for (int k = 0; k < KNN; ++k) {
    __builtin_nontemporal_store(__uint_as_float((unsigned)(keys[k] >> 32)),
                                o + (size_t)k * Ln);
    m = fmaxf(m, gather_g(xb, mb, y, xx, (int)(keys[k] & 31u)));
    if (STORE_KEYS)
      __builtin_nontemporal_store(keys[k],
                                  key_ws + ((size_t)b * KNN + k) * Ln + l);
  }

  __shared__ unsigned smax;
  if (threadIdx.x == 0) smax = 0u;
  __syncthreads();
  atomicMax(&smax, __float_as_uint(m));
  __syncthreads();
  if (threadIdx.x == 0) atomicMax(gmax, smax);
}

// Pass B: each wave owns 32 distinct pixels (two 16-column N-tiles). Every lane
// computes knn/gather/features for its own pixel and writes the zero-padded
// 20-float feature row to LDS (aligned b128 stores). After one barrier, both
// A (32x20 padded weights) and B (features) fragments are fetched by *address*
// with aligned ds_load_b64 -- no lane-variant register indexing, so no
// compare/cndmask LUTs. GEMM: V_WMMA_F32_16X16X4_F32, 5 k-steps x 2 M-tiles
// x 2 N-tiles = 20 WMMAs per wave; A fragments shared across N-tiles.
template <bool LOAD_KEYS>
__global__ void __launch_bounds__(256) wmma_pass_kernel(const float* __restrict__ x,
                                                        const int* __restrict__ mask,
                                                        const float* __restrict__ Wm,
                                                        const unsigned* __restrict__ gmaxp,
                                                        const u64* __restrict__ key_ws,
                                                        float* __restrict__ out) {
  __shared__ __align__(16) float lds_w[OUTC * KPAD];      // 2.5 KB padded weights
  __shared__ __align__(16) float lds_f[8 * 32 * KPAD];    // 20 KB: 8 waves x 32 px x 20
  for (int t = threadIdx.x; t < OUTC * KPAD; t += 256) {
    const int row = t / KPAD;
    const int k = t - row * KPAD;
    lds_w[t] = (k < FEATS) ? Wm[row * FEATS + k] : 0.f;
  }

  const int lane = threadIdx.x & 31;
  const int wave = threadIdx.x >> 5;
  const int grp = blockIdx.x * 8 + wave;    // 32-pixel group
  const int b = grp >> 12;                  // / (Ln/32) = /4096
  const int base = (grp & 4095) << 5;
  const int l = base + lane;                // unique pixel per lane
  const int y = l / Wn;
  const int xx = l - y * Wn;
  const float* xb = x + (size_t)b * Cn * Ln;
  const int*   mb = mask + (size_t)b * Ln;

  u64 keys[KNN];
  if (LOAD_KEYS) {
#pragma unroll
    for (int k = 0; k < KNN; ++k)
      keys[k] = key_ws[((size_t)b * KNN + k) * Ln + l];
  } else {
    knn_keys(xb, mb, y, xx, keys);
  }

  float g[KNN];
#pragma unroll
  for (int k = 0; k < KNN; ++k)
    g[k] = gather_g(xb, mb, y, xx, (int)(keys[k] & 31u));

  const float gmax = __uint_as_float(*gmaxp);
  const float range_self = fminf(fmaxf(g[0], 1.0f), gmax);
  const float cutoff = 0.05f * range_self;

  float feat[KPAD];
#pragma unroll
  for (int k = 0; k < KNN; ++k) {
    const float dist = __uint_as_float((unsigned)(keys[k] >> 32));
    feat[k] = (dist > cutoff) ? 0.f : g[k];
    feat[KNN + k] = (float)(int)(keys[k] & 31u);
  }
  feat[18] = 0.f;
  feat[19] = 0.f;

  // Lane-private feature row -> LDS (stride 80 B: 16-byte aligned).
  float* frow = &lds_f[(wave * 32 + lane) * KPAD];
#pragma unroll
  for (int s = 0; s < 5; ++s) {
    v4f t;
    t.x = feat[4 * s + 0];
    t.y = feat[4 * s + 1];
    t.z = feat[4 * s + 2];
    t.w = feat[4 * s + 3];
    *(v4f*)(frow + 4 * s) = t;
  }
  __syncthreads();  // covers weight staging + cross-lane feature visibility

  const bool hi = lane >= 16;
  const int n = lane & 15;                  // D column / A row within tile
  const int koff = hi ? 2 : 0;              // even -> 8-byte aligned b64 loads
  const float* wr0 = &lds_w[n * KPAD + koff];
  const float* wr1 = &lds_w[(n + 16) * KPAD + koff];
  const float* f0 = &lds_f[(wave * 32 + n) * KPAD + koff];        // N-tile 0
  const float* f1 = &lds_f[(wave * 32 + 16 + n) * KPAD + koff];   // N-tile 1

  v8f acc00 = {}, acc10 = {}, acc01 = {}, acc11 = {};
#pragma unroll
  for (int s = 0; s < 5; ++s) {
    const int k0 = 4 * s;
    const v2f a0 = *(const v2f*)(wr0 + k0);
    const v2f a1 = *(const v2f*)(wr1 + k0);
    const v2f b0 = *(const v2f*)(f0 + k0);
    const v2f b1 = *(const v2f*)(f1 + k0);
    acc00 = __builtin_amdgcn_wmma_f32_16x16x4_f32(false, a0, false, b0, (short)0, acc00, false, false);
    acc10 = __builtin_amdgcn_wmma_f32_16x16x4_f32(false, a1, false, b0, (short)0, acc10, false, false);
    acc01 = __builtin_amdgcn_wmma_f32_16x16x4_f32(false, a0, false, b1, (short)0, acc01, false, false);
    acc11 = __builtin_amdgcn_wmma_f32_16x16x4_f32(false, a1, false, b1, (short)0, acc11, false, false);
  }

  float* ob0 = out + (size_t)b * OUTC * Ln + base + n;        // N-tile 0 column
  float* ob1 = ob0 + 16;                                      // N-tile 1 column
  const int rbase = hi ? 8 : 0;
#pragma unroll
  for (int r = 0; r < 8; ++r) {
    float v00 = acc00[r], v10 = acc10[r], v01 = acc01[r], v11 = acc11[r];
    v00 = (v00 >= 0.f) ? v00 : LEAK * v00;
    v10 = (v10 >= 0.f) ? v10 : LEAK * v10;
    v01 = (v01 >= 0.f) ? v01 : LEAK * v01;
    v11 = (v11 >= 0.f) ? v11 : LEAK * v11;
    __builtin_nontemporal_store(v00, ob0 + (size_t)(rbase + r) * Ln);
    __builtin_nontemporal_store(v10, ob0 + (size_t)(rbase + r + 16) * Ln);
    __builtin_nontemporal_store(v01, ob1 + (size_t)(rbase + r) * Ln);
    __builtin_nontemporal_store(v11, ob1 + (size_t)(rbase + r + 16) * Ln);
  }
}

extern "C" void kernel_launch(void* const* d_in, const int* in_sizes, int n_in,
                              void* d_out, int out_size, void* d_ws, size_t ws_size,
                              hipStream_t stream) {
  (void)in_sizes; (void)n_in; (void)out_size;
  const float* x    = (const float*)d_in[0];
  const int*   mask = (const int*)d_in[1];   // bool mask delivered as int per harness rule
  const float* wgt  = (const float*)d_in[2]; // (32,2,3,3) == row-major (32,18)

  float* out     = (float*)d_out;                 // (4,32,64,2048)
  float* knn_out = out + (size_t)Bn * OUTC * Ln;  // (4,9,64,2048)

  unsigned* gmax = (unsigned*)d_ws;               // 4 bytes at ws[0]
  u64* key_ws = (u64*)((char*)d_ws + 256);        // sorted keys, 37.75 MB
  const size_t need = 256 + (size_t)Bn * KNN * Ln * sizeof(u64);
  const bool use_ws = ws_size >= need;

  init_gmax_kernel<<<1, 1, 0, stream>>>(gmax);
  if (use_ws) {
    knn_pass_kernel<true><<<(Bn * Ln) / 256, 256, 0, stream>>>(x, mask, knn_out, gmax, key_ws);
    wmma_pass_kernel<true><<<(Bn * Ln / 32) / 8, 256, 0, stream>>>(x, mask, wgt, gmax, key_ws, out);
  } else {
    knn_pass_kernel<false><<<(Bn * Ln) / 256, 256, 0, stream>>>(x, mask, knn_out, gmax, nullptr);
    wmma_pass_kernel<false><<<(Bn * Ln / 32) / 8, 256, 0, stream>>>(x, mask, wgt, gmax, nullptr, out);
  }
}